// Gat_55087250538921
// MI455X (gfx1250) — compile-verified
//
#include <hip/hip_runtime.h>
#include <hip/hip_bf16.h>

typedef __attribute__((ext_vector_type(2))) float v2f;
typedef __attribute__((ext_vector_type(8))) float v8f;

#define NNODES 100000
#define NEDGES 1600000

// ---------- helpers ----------
__device__ __forceinline__ float lrelu(float x) { return x > 0.0f ? x : 0.2f * x; }

// order-preserving float -> uint key (for atomicMax-based segment max)
__device__ __forceinline__ unsigned fkey(float f) {
    unsigned u = __float_as_uint(f);
    return (u & 0x80000000u) ? ~u : (u | 0x80000000u);
}
__device__ __forceinline__ float finv(unsigned k) {
    return (k & 0x80000000u) ? __uint_as_float(k ^ 0x80000000u) : __uint_as_float(~k);
}

// ---------- zero scratch ----------
__global__ void zero_kernel(float* __restrict__ p, int n) {
    int i = blockIdx.x * blockDim.x + threadIdx.x;
    int stride = gridDim.x * blockDim.x;
    for (; i < n; i += stride) p[i] = 0.0f;
}

// ---------- WMMA GEMM: H = pre(X) @ W,  X is [N,128], W is [128,HF] ----------
// pre(x) = relu(x + bias[c]) when FUSE (previous layer's bias+ReLU fused into load).
// One block = one 16-row tile; wave w computes 16-column tile w. HF/16 waves.
// Two independent accumulator chains to halve WMMA RAW dependency depth.
template <bool FUSE>
__global__ void gemm_wmma_kernel(const float* __restrict__ X, const float* __restrict__ W,
                                 const float* __restrict__ bias, float* __restrict__ Hout,
                                 int HF) {
    __shared__ float Xs[16 * 132];  // stride 132 -> conflict-free A-frag reads
    const int row0 = blockIdx.x * 16;
    const int tid = threadIdx.x;

    // stage 16x128 input tile (coalesced), applying fused relu(x+b) epilogue of prev layer
    for (int i = tid; i < 16 * 128; i += blockDim.x) {
        int r = i >> 7, c = i & 127;
        float v = X[(size_t)(row0 + r) * 128 + c];
        if (FUSE) v = fmaxf(v + bias[c], 0.0f);
        Xs[r * 132 + c] = v;
    }
    __syncthreads();

    const int wave = tid >> 5;
    const int lane = tid & 31;
    const int half = lane >> 4;   // 0: lanes 0-15, 1: lanes 16-31
    const int mn   = lane & 15;   // A row (M) and B col (N)
    const int col0 = wave * 16;
    const int kb   = half * 2;    // K sub-offset per ISA A/B layout

    v8f acc0 = {}, acc1 = {};
    for (int k0 = 0; k0 < 128; k0 += 8) {
        v2f a0, b0f, a1, b1f;
        a0.x = Xs[mn * 132 + k0 + kb];
        a0.y = Xs[mn * 132 + k0 + kb + 1];
        a1.x = Xs[mn * 132 + k0 + 4 + kb];
        a1.y = Xs[mn * 132 + k0 + 4 + kb + 1];
        const float* wp0 = W + (size_t)(k0 + kb) * HF + col0 + mn;
        const float* wp1 = W + (size_t)(k0 + 4 + kb) * HF + col0 + mn;
        b0f.x = wp0[0];
        b0f.y = wp0[HF];
        b1f.x = wp1[0];
        b1f.y = wp1[HF];
        acc0 = __builtin_amdgcn_wmma_f32_16x16x4_f32(false, a0, false, b0f, (short)0, acc0,
                                                     false, false);
        acc1 = __builtin_amdgcn_wmma_f32_16x16x4_f32(false, a1, false, b1f, (short)0, acc1,
                                                     false, false);
    }
    v8f acc = acc0 + acc1;
    // C/D layout: VGPR r holds M = r + 8*half, N = mn
#pragma unroll
    for (int r = 0; r < 8; ++r)
        Hout[(size_t)(row0 + half * 8 + r) * HF + col0 + mn] = acc[r];
}

// ---------- attention scalars: el[n,h] = <h[n,h,:], al[h,:]>, same for er ----------
__global__ void attn_kernel(const float* __restrict__ h, const float* __restrict__ al,
                            const float* __restrict__ ar, float* __restrict__ el,
                            float* __restrict__ er, int H, int F) {
    int idx = blockIdx.x * blockDim.x + threadIdx.x;
    if (idx >= NNODES * H) return;
    int node = idx / H, hd = idx % H;
    // all bases are 32-float (128B) aligned -> float4 loads
    const float4* hp  = (const float4*)(h + (size_t)node * H * F + hd * F);
    const float4* alp = (const float4*)(al + hd * F);
    const float4* arp = (const float4*)(ar + hd * F);
    float sl = 0.0f, sr = 0.0f;
    for (int f = 0; f < (F >> 2); ++f) {
        float4 v = hp[f], a = alp[f], b = arp[f];
        sl += v.x * a.x + v.y * a.y + v.z * a.z + v.w * a.w;
        sr += v.x * b.x + v.y * b.y + v.z * b.z + v.w * b.w;
    }
    el[idx] = sl;
    er[idx] = sr;
}

// ---------- edge pass 1: segment max (thread per edge, heads unrolled) ----------
template <int H>
__global__ void edge_max_kernel(const int* __restrict__ src, const int* __restrict__ dst,
                                const float* __restrict__ el, const float* __restrict__ er,
                                unsigned* __restrict__ mkey) {
    int e = blockIdx.x * blockDim.x + threadIdx.x;
    if (e >= NEDGES) return;
    int s = src[e], d = dst[e];
#pragma unroll
    for (int hd = 0; hd < H; ++hd) {
        float x = lrelu(el[s * H + hd] + er[d * H + hd]);
        atomicMax(&mkey[d * H + hd], fkey(x));
    }
}

// ---------- edge pass 2: den[dst,h] += exp(e - m) ----------
template <int H>
__global__ void edge_sum_kernel(const int* __restrict__ src, const int* __restrict__ dst,
                                const float* __restrict__ el, const float* __restrict__ er,
                                const unsigned* __restrict__ mkey, float* __restrict__ den) {
    int e = blockIdx.x * blockDim.x + threadIdx.x;
    if (e >= NEDGES) return;
    int s = src[e], d = dst[e];
#pragma unroll
    for (int hd = 0; hd < H; ++hd) {
        float x = lrelu(el[s * H + hd] + er[d * H + hd]);
        float m = finv(mkey[d * H + hd]);
        atomicAdd(&den[d * H + hd], __expf(x - m));
    }
}

// ---------- edge pass 3: obuf[dst,:] += h[src,:] * alpha  (one wave32 per edge) ----------
// lanes 0..H-1 compute per-head alpha once; all lanes fetch it via wave32 shuffle.
template <int H, int F, int HF>
__global__ void edge_scatter_kernel(const int* __restrict__ src, const int* __restrict__ dst,
                                    const float* __restrict__ hsrc, const float* __restrict__ el,
                                    const float* __restrict__ er, const unsigned* __restrict__ mkey,
                                    const float* __restrict__ den, float* __restrict__ obuf) {
    int wid  = (blockIdx.x * blockDim.x + threadIdx.x) >> 5;
    int lane = threadIdx.x & 31;
    if (wid >= NEDGES) return;
    int s = src[wid], d = dst[wid];
    float alpha_l = 0.0f;
    if (lane < H) {
        float x = lrelu(el[s * H + lane] + er[d * H + lane]);
        float m = finv(mkey[d * H + lane]);
        alpha_l = __expf(x - m) / den[d * H + lane];
    }
#pragma unroll
    for (int j = 0; j < (HF >> 5); ++j) {
        int c  = lane + 32 * j;
        int hd = c / F;  // compile-time shift
        float alpha = __shfl(alpha_l, hd, 32);
        atomicAdd(&obuf[(size_t)d * HF + c], hsrc[(size_t)s * HF + c] * alpha);
    }
}

// ---------- final: out = obuf + b2 (no relu on layer 2) ----------
__global__ void final_kernel(const float* __restrict__ obuf, const float* __restrict__ b,
                             float* __restrict__ out, int total, int C) {
    int idx = blockIdx.x * blockDim.x + threadIdx.x;
    if (idx >= total) return;
    out[idx] = obuf[idx] + b[idx % C];
}

static inline int cdiv(int a, int b) { return (a + b - 1) / b; }

extern "C" void kernel_launch(void* const* d_in, const int* in_sizes, int n_in,
                              void* d_out, int out_size, void* d_ws, size_t ws_size,
                              hipStream_t stream) {
    const float* feat = (const float*)d_in[0];
    const int*   src  = (const int*)d_in[1];
    const int*   dst  = (const int*)d_in[2];
    const float* W0   = (const float*)d_in[3];
    const float* al0  = (const float*)d_in[4];
    const float* ar0  = (const float*)d_in[5];
    const float* b0   = (const float*)d_in[6];
    const float* W1   = (const float*)d_in[7];
    const float* al1  = (const float*)d_in[8];
    const float* ar1  = (const float*)d_in[9];
    const float* b1   = (const float*)d_in[10];
    const float* W2   = (const float*)d_in[11];
    const float* al2  = (const float*)d_in[12];
    const float* ar2  = (const float*)d_in[13];
    const float* b2   = (const float*)d_in[14];

    // workspace layout (floats): hbuf[N*128] | obuf[N*128] | el[N*4] | er[N*4] | mkey[N*4] | den[N*4]
    float*    hbuf = (float*)d_ws;
    float*    obuf = hbuf + (size_t)NNODES * 128;
    float*    el   = obuf + (size_t)NNODES * 128;
    float*    er   = el + (size_t)NNODES * 4;
    unsigned* mkey = (unsigned*)(er + (size_t)NNODES * 4);
    float*    den  = (float*)(mkey + (size_t)NNODES * 4);

    const int rowTiles = NNODES / 16;  // 6250, exact

    for (int li = 0; li < 3; ++li) {
        const int H  = (li == 2) ? 1 : 4;
        const int F  = (li == 2) ? 64 : 32;
        const int HF = H * F;
        const float* W  = (li == 0) ? W0 : (li == 1) ? W1 : W2;
        const float* al = (li == 0) ? al0 : (li == 1) ? al1 : al2;
        const float* ar = (li == 0) ? ar0 : (li == 1) ? ar1 : ar2;
        const float* pb = (li == 1) ? b0 : b1;  // unused for li==0
        const float* X  = (li == 0) ? feat : obuf;
        const int gemm_threads = 32 * (HF / 16);

        // 1) h = pre(X) @ W  (WMMA f32 16x16x4, fused relu(x + prev_b) on input for li>0)
        if (li == 0)
            gemm_wmma_kernel<false><<<rowTiles, gemm_threads, 0, stream>>>(X, W, nullptr, hbuf, HF);
        else
            gemm_wmma_kernel<true><<<rowTiles, gemm_threads, 0, stream>>>(X, W, pb, hbuf, HF);

        // 2) zero accumulators for this layer (after GEMM consumed obuf); mkey & den adjacent
        zero_kernel<<<2048, 256, 0, stream>>>(obuf, NNODES * HF);
        zero_kernel<<<512, 256, 0, stream>>>((float*)mkey, 2 * NNODES * H);

        // 3) attention scalars
        attn_kernel<<<cdiv(NNODES * H, 256), 256, 0, stream>>>(hbuf, al, ar, el, er, H, F);

        // 4-6) edge phase (hbuf + obuf ~102 MB, resident in the 192 MB L2)
        if (H == 4) {
            edge_max_kernel<4><<<cdiv(NEDGES, 256), 256, 0, stream>>>(src, dst, el, er, mkey);
            edge_sum_kernel<4><<<cdiv(NEDGES, 256), 256, 0, stream>>>(src, dst, el, er, mkey, den);
            edge_scatter_kernel<4, 32, 128><<<cdiv(NEDGES, 8), 256, 0, stream>>>(
                src, dst, hbuf, el, er, mkey, den, obuf);
        } else {
            edge_max_kernel<1><<<cdiv(NEDGES, 256), 256, 0, stream>>>(src, dst, el, er, mkey);
            edge_sum_kernel<1><<<cdiv(NEDGES, 256), 256, 0, stream>>>(src, dst, el, er, mkey, den);
            edge_scatter_kernel<1, 64, 64><<<cdiv(NEDGES, 8), 256, 0, stream>>>(
                src, dst, hbuf, el, er, mkey, den, obuf);
        }
    }

    // layer 2 output: out = obuf + b2 (no activation)
    final_kernel<<<cdiv(NNODES * 64, 256), 256, 0, stream>>>(obuf, b2, (float*)d_out, NNODES * 64, 64);
}